// BertGTHead_2482491097873
// MI455X (gfx1250) — compile-verified
//
#include <hip/hip_runtime.h>
#include <hip/hip_bf16.h>

typedef __attribute__((ext_vector_type(2))) float v2f;
typedef __attribute__((ext_vector_type(8))) float v8f;

#define BB    16
#define SS    512
#define HH    1024
#define GG    16
#define WIN   15
#define NTILE 64          // HH / 16
#define TPB   256         // 8 waves of 32

// ---------------------------------------------------------------------------
// Kernel 1: one block per (batch b, 16-wide H tile).
// Streams the 512x16 f32 slice into LDS, builds the 512x16 window-mask tile
// in LDS, then computes:
//   - window avg numerators P[g,h] via V_WMMA_F32_16X16X4_F32 (K split over 8 waves)
//   - window max, num_tokens, gap gather
//   - text max / text sum / num_t
// and reduces to per-(b,tile) partial dot products against W_gap / W_cls.
// ---------------------------------------------------------------------------
__global__ __launch_bounds__(TPB)
void bertgt_pool_kernel(const float* __restrict__ seq,
                        const float* __restrict__ pooled,
                        const int*   __restrict__ ttids,
                        const int*   __restrict__ wordmask,
                        const int*   __restrict__ gapids,
                        const float* __restrict__ Wgap,
                        const float* __restrict__ Wcls,
                        float*       __restrict__ ws_cls,
                        float*       __restrict__ ws_gap)
{
    __shared__ float sSeq[SS][16];        // 32 KB seq slice  (row s, col h)
    __shared__ float sWmk[SS][16];        // 32 KB wmask tile (row s, col g)
    __shared__ float sBase[SS];           // base mask as float
    __shared__ float sPpart[8][16][16];   // per-wave WMMA partial tiles
    __shared__ int   sGid[GG];
    __shared__ float sGapRed[16][16];
    __shared__ float sTMax[16][16];
    __shared__ float sTSum[16][16];
    __shared__ float sNumT;

    const int tid  = threadIdx.x;
    const int b    = blockIdx.y;
    const int tile = blockIdx.x;
    const int n0   = tile * 16;

    if (tid < GG) sGid[tid] = gapids[b * GG + tid];

    for (int s = tid; s < SS; s += TPB) {
        const int tt = ttids[b * SS + s];
        const int wm = wordmask[b * SS + s];
        sBase[s] = (tt == 0 && wm != 0) ? 1.0f : 0.0f;
    }
    __syncthreads();

    // Load seq slice and build wmask tile: wmask[s][g] = base[s] * inwindow(g,s).
    const float* seqb = seq + ((size_t)b * SS) * HH + n0;
    #pragma unroll
    for (int i = 0; i < 32; ++i) {
        const int e = tid + i * TPB;
        const int s = e >> 4;
        const int c = e & 15;
        sSeq[s][c] = seqb[(size_t)s * HH + c];
        const int gi = sGid[c];
        const float inwin = ((unsigned)(s - (gi - WIN)) <= (unsigned)(2 * WIN)) ? 1.0f : 0.0f;
        sWmk[s][c] = sBase[s] * inwin;
    }
    __syncthreads();

    // ---- WMMA: P[g,h] = sum_s wmask[g,s] * seq[s,h]; K=512 split over 8 waves ----
    {
        const int wave = tid >> 5;
        const int lane = tid & 31;
        const int mn   = lane & 15;   // A: row m (gap g) / B: col n (h)
        const int kh   = lane >> 4;   // which K pair this half-wave holds
        v8f acc = {0.f, 0.f, 0.f, 0.f, 0.f, 0.f, 0.f, 0.f};
        const int k0base = wave * 64;
        #pragma unroll
        for (int kk = 0; kk < 64; kk += 4) {
            const int s0 = k0base + kk + 2 * kh;
            const int s1 = s0 + 1;
            v2f av, bv;
            av.x = sWmk[s0][mn];
            av.y = sWmk[s1][mn];
            bv.x = sSeq[s0][mn];
            bv.y = sSeq[s1][mn];
            acc = __builtin_amdgcn_wmma_f32_16x16x4_f32(
                false, av, false, bv, (short)0, acc, false, false);
        }
        #pragma unroll
        for (int i = 0; i < 8; ++i)
            sPpart[wave][i + 8 * kh][mn] = acc[i];
    }
    __syncthreads();

    const int g = tid >> 4;   // also the 32-row chunk index for text pools
    const int h = tid & 15;

    // Deterministic reduction of the 8 per-wave partial tiles.
    float P = 0.0f;
    #pragma unroll
    for (int w = 0; w < 8; ++w) P += sPpart[w][g][h];

    const int gi = sGid[g];
    const int lo = (gi - WIN < 0) ? 0 : gi - WIN;
    const int hi = (gi + WIN > SS - 1) ? SS - 1 : gi + WIN;

    // num_tokens[g] and window max (mask-multiply semantics: zeros participate,
    // and out-of-window zeros always exist since 2*WIN+1 << SS).
    float nt   = 0.0f;
    float wmax = 0.0f;
    for (int s = lo; s <= hi; ++s) {
        const float bs = sBase[s];
        nt += bs;
        wmax = fmaxf(wmax, sSeq[s][h] * bs);
    }

    const float gapval = sSeq[gi][h];
    sGapRed[g][h] = gapval * Wgap[n0 + h]
                  + wmax   * Wgap[HH + n0 + h]
                  + (P / nt) * Wgap[2 * HH + n0 + h];

    // Text pools: chunk g covers rows [32g, 32g+32) for column h.
    float tmax = -INFINITY, tsum = 0.0f;
    const int sbeg = g * 32;
    for (int s = sbeg; s < sbeg + 32; ++s) {
        const float v = sSeq[s][h] * sBase[s];
        tmax = fmaxf(tmax, v);
        tsum += v;
    }
    sTMax[g][h] = tmax;
    sTSum[g][h] = tsum;

    if (tid == 0) {
        float c = 0.0f;
        for (int s = 0; s < SS; ++s) c += sBase[s];
        sNumT = c;
    }
    __syncthreads();

    if (h == 0) {
        float acc = 0.0f;
        #pragma unroll
        for (int j = 0; j < 16; ++j) acc += sGapRed[g][j];
        ws_gap[((size_t)b * NTILE + tile) * GG + g] = acc;
    }
    if (tid == 0) {
        const float invn = 1.0f / sNumT;
        float pcls = 0.0f;
        for (int j = 0; j < 16; ++j) {
            float tm = sTMax[0][j];
            float ts = sTSum[0][j];
            for (int c = 1; c < 16; ++c) {
                tm = fmaxf(tm, sTMax[c][j]);
                ts += sTSum[c][j];
            }
            pcls += pooled[(size_t)b * HH + n0 + j] * Wcls[n0 + j]
                  + tm * Wcls[HH + n0 + j]
                  + ts * invn * Wcls[2 * HH + n0 + j];
        }
        ws_cls[(size_t)b * NTILE + tile] = pcls;
    }
}

// ---------------------------------------------------------------------------
// Kernel 2: deterministic reduction of per-tile partials -> (B,17) output.
// ---------------------------------------------------------------------------
__global__ __launch_bounds__(32)
void bertgt_reduce_kernel(const float* __restrict__ ws_cls,
                          const float* __restrict__ ws_gap,
                          const float* __restrict__ bgap,
                          const float* __restrict__ bcls,
                          float* __restrict__ out)
{
    const int b = blockIdx.x;
    const int t = threadIdx.x;
    if (t == 0) {
        float acc = bcls[0];
        for (int n = 0; n < NTILE; ++n) acc += ws_cls[(size_t)b * NTILE + n];
        out[b * 17] = acc;
    } else if (t <= GG) {
        const int g = t - 1;
        float acc = bgap[0];
        for (int n = 0; n < NTILE; ++n)
            acc += ws_gap[((size_t)b * NTILE + n) * GG + g];
        out[b * 17 + t] = acc;
    }
}

extern "C" void kernel_launch(void* const* d_in, const int* in_sizes, int n_in,
                              void* d_out, int out_size, void* d_ws, size_t ws_size,
                              hipStream_t stream) {
    (void)in_sizes; (void)n_in; (void)out_size; (void)ws_size;
    const float* seq    = (const float*)d_in[0];
    const float* pooled = (const float*)d_in[1];
    const int*   ttids  = (const int*)d_in[2];
    const int*   wmask  = (const int*)d_in[3];
    const int*   gapids = (const int*)d_in[4];
    const float* Wgap   = (const float*)d_in[5];
    const float* bgap   = (const float*)d_in[6];
    const float* Wcls   = (const float*)d_in[7];
    const float* bcls   = (const float*)d_in[8];
    float* out = (float*)d_out;

    float* ws_cls = (float*)d_ws;                 // BB*NTILE floats
    float* ws_gap = ws_cls + (size_t)BB * NTILE;  // BB*NTILE*GG floats

    dim3 grid(NTILE, BB);
    bertgt_pool_kernel<<<grid, TPB, 0, stream>>>(seq, pooled, ttids, wmask, gapids,
                                                 Wgap, Wcls, ws_cls, ws_gap);
    bertgt_reduce_kernel<<<BB, 32, 0, stream>>>(ws_cls, ws_gap, bgap, bcls, out);
}